// GMNLayerX_Pooling3_28432683499986
// MI455X (gfx1250) — compile-verified
//
#include <hip/hip_runtime.h>
#include <math.h>

#define TT 10
#define NN 200000
#define CC 4
#define SS 100000

typedef __attribute__((ext_vector_type(2))) float v2f;
typedef __attribute__((ext_vector_type(8))) float v8f;

// ---------------------------------------------------------------------------
// ws layout (floats): [0..15] theta-coefs (padded, x_last folded in)
//                     [16..31] softmax(time_weights) (padded)
//                     [32 .. 32+N*12)   omega accumulator (N,C,3)
//                     [32+N*12 .. +N)   counts
// ---------------------------------------------------------------------------

__global__ void k_setup(const float* __restrict__ theta,
                        const float* __restrict__ tw,
                        float* __restrict__ wsf) {
  if (blockIdx.x == 0 && threadIdx.x == 0) {
    // softmax over time_weights
    float m = tw[0];
    for (int t = 1; t < TT; ++t) m = fmaxf(m, tw[t]);
    float e[TT];
    float se = 0.f;
    for (int t = 0; t < TT; ++t) { e[t] = expf(tw[t] - m); se += e[t]; }
    float inv = 1.f / se;
    for (int t = 0; t < 16; ++t) wsf[16 + t] = (t < TT) ? e[t] * inv : 0.f;
    // theta coefs with x_last folded in:  out = sum_t coef[t] * x[t]
    float ssum = 0.f;
    for (int t = 0; t < TT; ++t) ssum += theta[t];
    for (int t = 0; t < 16; ++t) {
      float cv = (t < TT) ? theta[t] : 0.f;
      if (t == TT - 1) cv += 1.f - ssum;
      wsf[t] = cv;
    }
  }
}

__global__ void k_zero(float* __restrict__ p, size_t n) {
  size_t i = (size_t)blockIdx.x * blockDim.x + threadIdx.x;
  size_t stride = (size_t)gridDim.x * blockDim.x;
  for (; i < n; i += stride) p[i] = 0.f;
}

// One thread per stick: accumulate softmax-weighted omega into per-node slots.
// omega2 == omega1 (cross(-r,-v) = cross(r,v), |r2| = |r1|), so compute once.
__global__ void k_sticks(const float* __restrict__ x,
                         const float* __restrict__ vec,
                         const int* __restrict__ sc,
                         const float* __restrict__ wsf,
                         float* __restrict__ acc,
                         float* __restrict__ cnt) {
  int s = blockIdx.x * blockDim.x + threadIdx.x;
  if (s >= SS) return;
  int n1 = sc[2 * s];
  int n2 = sc[2 * s + 1];

  float w[TT];
#pragma unroll
  for (int t = 0; t < TT; ++t) w[t] = wsf[16 + t];

  float osum[12];
#pragma unroll
  for (int i = 0; i < 12; ++i) osum[i] = 0.f;

  for (int t = 0; t < TT; ++t) {
    const float4* p1 = reinterpret_cast<const float4*>(x + ((size_t)t * NN + n1) * 12);
    const float4* p2 = reinterpret_cast<const float4*>(x + ((size_t)t * NN + n2) * 12);
    const float4* q1 = reinterpret_cast<const float4*>(vec + ((size_t)t * NN + n1) * 12);
    const float4* q2 = reinterpret_cast<const float4*>(vec + ((size_t)t * NN + n2) * 12);
    float4 A0 = p1[0], A1 = p1[1], A2 = p1[2];
    float4 B0 = p2[0], B1 = p2[1], B2 = p2[2];
    float4 U0 = q1[0], U1 = q1[1], U2 = q1[2];
    float4 V0 = q2[0], V1 = q2[1], V2 = q2[2];
    float x1[12] = {A0.x, A0.y, A0.z, A0.w, A1.x, A1.y, A1.z, A1.w, A2.x, A2.y, A2.z, A2.w};
    float x2[12] = {B0.x, B0.y, B0.z, B0.w, B1.x, B1.y, B1.z, B1.w, B2.x, B2.y, B2.z, B2.w};
    float v1[12] = {U0.x, U0.y, U0.z, U0.w, U1.x, U1.y, U1.z, U1.w, U2.x, U2.y, U2.z, U2.w};
    float v2[12] = {V0.x, V0.y, V0.z, V0.w, V1.x, V1.y, V1.z, V1.w, V2.x, V2.y, V2.z, V2.w};
    float wt = w[t];
#pragma unroll
    for (int c = 0; c < CC; ++c) {
      float rx = 0.5f * (x1[c * 3 + 0] - x2[c * 3 + 0]);
      float ry = 0.5f * (x1[c * 3 + 1] - x2[c * 3 + 1]);
      float rz = 0.5f * (x1[c * 3 + 2] - x2[c * 3 + 2]);
      float vx = 0.5f * (v1[c * 3 + 0] - v2[c * 3 + 0]);
      float vy = 0.5f * (v1[c * 3 + 1] - v2[c * 3 + 1]);
      float vz = 0.5f * (v1[c * 3 + 2] - v2[c * 3 + 2]);
      float cx = ry * vz - rz * vy;
      float cy = rz * vx - rx * vz;
      float cz = rx * vy - ry * vx;
      float rn = sqrtf(rx * rx + ry * ry + rz * rz);
      rn = fmaxf(rn, 1e-3f);
      float ws = wt / (rn * rn);
      osum[c * 3 + 0] += cx * ws;
      osum[c * 3 + 1] += cy * ws;
      osum[c * 3 + 2] += cz * ws;
    }
  }
  float* a1 = acc + (size_t)n1 * 12;
  float* a2 = acc + (size_t)n2 * 12;
#pragma unroll
  for (int i = 0; i < 12; ++i) {
    atomicAdd(a1 + i, osum[i]);
    atomicAdd(a2 + i, osum[i]);
  }
  atomicAdd(cnt + n1, 1.f);
  atomicAdd(cnt + n2, 1.f);
}

// Theta contraction:  out[m] = sum_{t<10} coef[t] * x[t, m]   (m = flat n*12+c*3+d)
// Done as V_WMMA_F32_16X16X4_F32, 3 K-tiles (T padded 10 -> 12 with zero coefs).
// A (16x4 f32): VGPR0 = K0(lanes0-15)/K2(lanes16-31), VGPR1 = K1/K3; M = lane&15.
// B (4x16):     rows constant = coef[k], mirrored layout.
// D (16x16):    VGPR r = row M=r (lanes0-15) / M=8+r (lanes16-31); columns identical.
__global__ void k_theta_wmma(const float* __restrict__ x,
                             const float* __restrict__ wsf,
                             float* __restrict__ out) {
  const size_t M = (size_t)NN * CC * 3;  // 2,400,000 -> 150,000 tiles of 16
  const int ntiles = (int)(M / 16);
  int lane = threadIdx.x & 31;
  int half = lane >> 4;
  int lm = lane & 15;
  int wib = threadIdx.x >> 5;
  int wpb = blockDim.x >> 5;
  int gw = blockIdx.x * wpb + wib;
  int nw = gridDim.x * wpb;

  for (int tile = gw; tile < ntiles; tile += nw) {  // wave-uniform: EXEC all-ones at WMMA
    size_t m0 = (size_t)tile * 16;
    v8f c = {0.f, 0.f, 0.f, 0.f, 0.f, 0.f, 0.f, 0.f};
#pragma unroll
    for (int kt = 0; kt < 3; ++kt) {
      int t0 = kt * 4 + 2 * half;
      int t1 = t0 + 1;
      int a0 = t0 < TT ? t0 : TT - 1;  // clamp address; coef is 0 there
      int a1 = t1 < TT ? t1 : TT - 1;
      v2f A, B;
      A.x = x[(size_t)a0 * M + m0 + lm];
      A.y = x[(size_t)a1 * M + m0 + lm];
      B.x = wsf[t0];
      B.y = wsf[t1];
      c = __builtin_amdgcn_wmma_f32_16x16x4_f32(false, A, false, B, (short)0, c,
                                                false, false);
    }
    if (lm == 0) {  // lane 0 -> rows 0..7, lane 16 -> rows 8..15
      float4 lo, hi;
      lo.x = c[0]; lo.y = c[1]; lo.z = c[2]; lo.w = c[3];
      hi.x = c[4]; hi.y = c[5]; hi.z = c[6]; hi.w = c[7];
      float* base = out + m0 + (size_t)(8 * half);
      *reinterpret_cast<float4*>(base) = lo;
      *reinterpret_cast<float4*>(base + 4) = hi;
    }
  }
}

__device__ __forceinline__ void rot_apply(const float o[3], const float r[3],
                                          float dst[3]) {
  float th = sqrtf(o[0] * o[0] + o[1] * o[1] + o[2] * o[2]);
  if (th < 1e-6f) {  // reference: R = I when ||omega|| < 1e-6
    dst[0] = r[0]; dst[1] = r[1]; dst[2] = r[2];
    return;
  }
  float inv = 1.f / th;
  float ux = o[0] * inv, uy = o[1] * inv, uz = o[2] * inv;
  float sn = sinf(th);
  float om = 1.f - cosf(th);
  float k1x = uy * r[2] - uz * r[1];
  float k1y = uz * r[0] - ux * r[2];
  float k1z = ux * r[1] - uy * r[0];
  float k2x = uy * k1z - uz * k1y;
  float k2y = uz * k1x - ux * k1z;
  float k2z = ux * k1y - uy * k1x;
  dst[0] = r[0] + sn * k1x + om * k2x;
  dst[1] = r[1] + sn * k1y + om * k2y;
  dst[2] = r[2] + sn * k1z + om * k2z;
}

// One thread per stick: Rodrigues-rotate half-sticks around midpoint, overwrite out.
__global__ void k_rotate(const float* __restrict__ x,
                         const int* __restrict__ sc,
                         const float* __restrict__ acc,
                         const float* __restrict__ cnt,
                         float* __restrict__ out) {
  int s = blockIdx.x * blockDim.x + threadIdx.x;
  if (s >= SS) return;
  int n1 = sc[2 * s];
  int n2 = sc[2 * s + 1];
  const float* xl = x + (size_t)(TT - 1) * NN * 12;  // x_last
  const float4* p1 = reinterpret_cast<const float4*>(xl + (size_t)n1 * 12);
  const float4* p2 = reinterpret_cast<const float4*>(xl + (size_t)n2 * 12);
  float4 A0 = p1[0], A1 = p1[1], A2 = p1[2];
  float4 B0 = p2[0], B1 = p2[1], B2 = p2[2];
  float x1[12] = {A0.x, A0.y, A0.z, A0.w, A1.x, A1.y, A1.z, A1.w, A2.x, A2.y, A2.z, A2.w};
  float x2[12] = {B0.x, B0.y, B0.z, B0.w, B1.x, B1.y, B1.z, B1.w, B2.x, B2.y, B2.z, B2.w};
  float i1 = 1.f / fmaxf(cnt[n1], 1.f);
  float i2 = 1.f / fmaxf(cnt[n2], 1.f);
  const float* o1p = acc + (size_t)n1 * 12;
  const float* o2p = acc + (size_t)n2 * 12;
  float out1[12], out2[12];
#pragma unroll
  for (int c = 0; c < CC; ++c) {
    float xc[3], r1[3], r2[3], o1[3], o2[3], t1[3], t2[3];
#pragma unroll
    for (int d = 0; d < 3; ++d) {
      xc[d] = 0.5f * (x1[c * 3 + d] + x2[c * 3 + d]);
      r1[d] = x1[c * 3 + d] - xc[d];
      r2[d] = -r1[d];
      o1[d] = o1p[c * 3 + d] * i1;
      o2[d] = o2p[c * 3 + d] * i2;
    }
    rot_apply(o1, r1, t1);
    rot_apply(o2, r2, t2);
#pragma unroll
    for (int d = 0; d < 3; ++d) {
      out1[c * 3 + d] = xc[d] + t1[d];
      out2[c * 3 + d] = xc[d] + t2[d];
    }
  }
  float4* w1 = reinterpret_cast<float4*>(out + (size_t)n1 * 12);
  float4* w2 = reinterpret_cast<float4*>(out + (size_t)n2 * 12);
  w1[0] = make_float4(out1[0], out1[1], out1[2], out1[3]);
  w1[1] = make_float4(out1[4], out1[5], out1[6], out1[7]);
  w1[2] = make_float4(out1[8], out1[9], out1[10], out1[11]);
  w2[0] = make_float4(out2[0], out2[1], out2[2], out2[3]);
  w2[1] = make_float4(out2[4], out2[5], out2[6], out2[7]);
  w2[2] = make_float4(out2[8], out2[9], out2[10], out2[11]);
}

extern "C" void kernel_launch(void* const* d_in, const int* in_sizes, int n_in,
                              void* d_out, int out_size, void* d_ws, size_t ws_size,
                              hipStream_t stream) {
  // inputs: 0:h (unused) 1:edge_index (unused) 2:x 3:vec 4:stick_config 5:theta 6:time_weights
  const float* x   = (const float*)d_in[2];
  const float* vec = (const float*)d_in[3];
  const int*   sc  = (const int*)d_in[4];
  const float* th  = (const float*)d_in[5];
  const float* tw  = (const float*)d_in[6];
  float* out = (float*)d_out;

  float* wsf = (float*)d_ws;
  float* acc = wsf + 32;                       // N*12 floats
  float* cnt = acc + (size_t)NN * 12;          // N floats

  k_setup<<<1, 32, 0, stream>>>(th, tw, wsf);

  size_t zn = (size_t)NN * 12 + (size_t)NN;    // acc + cnt contiguous
  k_zero<<<2048, 256, 0, stream>>>(acc, zn);

  k_sticks<<<(SS + 255) / 256, 256, 0, stream>>>(x, vec, sc, wsf, acc, cnt);

  const int ntiles = (NN * CC * 3) / 16;       // 150,000
  const int wpb = 8;                           // 256 threads = 8 wave32
  int blocks = (ntiles + wpb - 1) / wpb;
  k_theta_wmma<<<blocks, 256, 0, stream>>>(x, wsf, out);

  k_rotate<<<(SS + 255) / 256, 256, 0, stream>>>(x, sc, acc, cnt, out);
}